// MyModel_61933428409288
// MI455X (gfx1250) — compile-verified
//
#include <hip/hip_runtime.h>
#include <cstdint>

// out[i] = max{ j : mapping[j] == tensor[i] }, else 0.
// Values in [0, 1024) -> 1024-entry LDS LUT built with ds_max atomics.
// Mapping staged into LDS with gfx1250 GLOBAL_LOAD_ASYNC_TO_LDS_B32 (ASYNCcnt).
// Streaming gather: int4 loads, float4 stores (HBM-bound, ~8 MB total traffic).

#define LUT_SIZE 1024
#define ELEMS_PER_BLOCK 1024  // 256 threads x 4 elements

__global__ __launch_bounds__(256) void lut_gather_kernel(
    const int* __restrict__ tensor,
    const int* __restrict__ mapping,
    float* __restrict__ out,
    int n, int m)
{
    __shared__ int lut[LUT_SIZE];
    __shared__ int map_lds[512];

    const int tid = threadIdx.x;
    const long long base = (long long)blockIdx.x * ELEMS_PER_BLOCK + (long long)tid * 4;

    // Warm this block's tensor slice while we build the LUT (global_prefetch_b8).
    if (base < n) __builtin_prefetch(tensor + base, 0, 0);

    // Zero-init LUT (ds_store). 0 == "no match" == "match at j=0" -> exact.
    lut[tid]       = 0;
    lut[tid + 256] = 0;
    lut[tid + 512] = 0;
    lut[tid + 768] = 0;

    // Async-copy mapping -> LDS: per-lane GLOBAL_LOAD_ASYNC_TO_LDS_B32 in GVS
    // mode (saddr = mapping base, vaddr = byte offset, vdst = LDS byte addr).
    // Low 32 bits of a flat shared-space pointer are the wave-relative LDS
    // offset (ISA 10.2).
    {
        const uint32_t lds_base = (uint32_t)(uintptr_t)(&map_lds[0]);
        if (tid < m) {
            uint32_t laddr = lds_base + 4u * (uint32_t)tid;
            uint32_t goff  = 4u * (uint32_t)tid;
            asm volatile("global_load_async_to_lds_b32 %0, %1, %2"
                         :: "v"(laddr), "v"(goff), "s"(mapping) : "memory");
        }
        if (tid + 256 < m) {
            uint32_t laddr = lds_base + 4u * (uint32_t)(tid + 256);
            uint32_t goff  = 4u * (uint32_t)(tid + 256);
            asm volatile("global_load_async_to_lds_b32 %0, %1, %2"
                         :: "v"(laddr), "v"(goff), "s"(mapping) : "memory");
        }
        asm volatile("s_wait_asynccnt 0" ::: "memory");
    }

    __syncthreads();

    // Build LUT: lut[v] = max j with mapping[j]==v  (ds_max_i32).
    if (tid < m) {
        int v = map_lds[tid];
        atomicMax(&lut[v & (LUT_SIZE - 1)], tid);
    }
    if (tid + 256 < m) {
        int v = map_lds[tid + 256];
        atomicMax(&lut[v & (LUT_SIZE - 1)], tid + 256);
    }

    __syncthreads();

    // Streaming gather: 16B loads, 16B stores.
    if (base + 3 < n) {
        int4 v = *(const int4*)(tensor + base);
        float4 r;
        r.x = (float)lut[v.x & (LUT_SIZE - 1)];
        r.y = (float)lut[v.y & (LUT_SIZE - 1)];
        r.z = (float)lut[v.z & (LUT_SIZE - 1)];
        r.w = (float)lut[v.w & (LUT_SIZE - 1)];
        *(float4*)(out + base) = r;
    } else {
        for (int k = 0; k < 4; ++k) {
            long long i = base + k;
            if (i < n) out[i] = (float)lut[tensor[i] & (LUT_SIZE - 1)];
        }
    }
}

extern "C" void kernel_launch(void* const* d_in, const int* in_sizes, int n_in,
                              void* d_out, int out_size, void* d_ws, size_t ws_size,
                              hipStream_t stream) {
    (void)n_in; (void)d_ws; (void)ws_size;
    const int* tensor  = (const int*)d_in[0];
    const int* mapping = (const int*)d_in[1];
    float* out = (float*)d_out;

    const int n = in_sizes[0];   // 1,048,576
    const int m = in_sizes[1];   // 512

    const int blocks = (n + ELEMS_PER_BLOCK - 1) / ELEMS_PER_BLOCK;
    lut_gather_kernel<<<blocks, 256, 0, stream>>>(tensor, mapping, out, n, m);
}